// GraphSAGE_44925357916337
// MI455X (gfx1250) — compile-verified
//
#include <hip/hip_runtime.h>

typedef __attribute__((ext_vector_type(2))) float v2f;
typedef __attribute__((ext_vector_type(8))) float v8f;

#define IN_CH   128
#define HID_CH  256
#define OUT_CH  128
#define N_GRAPHS 64
#define KC      64   // K-chunk staged in LDS

// ---------------- degree / reciprocal ----------------
__global__ void deg_kernel(const int* __restrict__ dst, float* __restrict__ cnt, int E) {
    int e = blockIdx.x * blockDim.x + threadIdx.x;
    if (e < E) atomicAdd(&cnt[dst[e]], 1.0f);
}

__global__ void inv_kernel(const float* __restrict__ cnt, float* __restrict__ inv, int N) {
    int i = blockIdx.x * blockDim.x + threadIdx.x;
    if (i < N) inv[i] = 1.0f / fmaxf(cnt[i], 1.0f);
}

// ---------------- weight transpose: W[Out][In] -> Wt[In][Out] ----------------
__global__ void transpose_kernel(const float* __restrict__ W, float* __restrict__ Wt,
                                 int Out, int In) {
    int t = blockIdx.x * blockDim.x + threadIdx.x;
    if (t < Out * In) {
        int o = t / In, k = t - o * In;
        Wt[k * Out + o] = W[t];
    }
}

// ---------------- edge scatter-add: agg[dst] += feat[src] ----------------
__global__ void scatter_kernel(const float* __restrict__ feat,
                               const int* __restrict__ src, const int* __restrict__ dst,
                               float* __restrict__ agg, int E, int C) {
    int t = blockIdx.x * blockDim.x + threadIdx.x;
    int chunks = C >> 2;
    int e = t / chunks;
    if (e >= E) return;
    int ch = (t - e * chunks) << 2;
    int s = src[e], d = dst[e];
    const float4 v = *(const float4*)(feat + (size_t)s * C + ch);
    float* o = agg + (size_t)d * C + ch;
    atomicAdd(o + 0, v.x);
    atomicAdd(o + 1, v.y);
    atomicAdd(o + 2, v.z);
    atomicAdd(o + 3, v.w);
}

// ---------------- fused SAGE GEMM:  out = (A0 .* inv) @ B0 + A1 @ B1 + bias ----------------
// B0/B1 pre-transposed to [K][OutC]. Block: 256 thr = 8 waves, tile 32(M) x 64(N).
// WMMA f32 16x16x4, K staged through LDS in chunks of KC.
__global__ __launch_bounds__(256)
void sage_gemm_kernel(const float* __restrict__ A0, const float* __restrict__ A1,
                      const float* __restrict__ B0, const float* __restrict__ B1,
                      const float* __restrict__ bias, const float* __restrict__ inv,
                      float* __restrict__ out, int M, int K, int OutC, int do_relu) {
    __shared__ float As[32][KC + 1];
    __shared__ float Bs[KC][64 + 1];

    const int tid  = threadIdx.x;
    const int lane = tid & 31;
    const int wid  = tid >> 5;       // 0..7
    const int wrow = wid >> 2;       // 0..1 -> +16 rows
    const int wcol = wid & 3;        // 0..3 -> +16 cols
    const int ln   = lane & 15;
    const int hh   = lane >> 4;      // half-wave select

    const int m0 = blockIdx.x * 32;
    const int n0 = blockIdx.y * 64;

    v8f acc = {0.f, 0.f, 0.f, 0.f, 0.f, 0.f, 0.f, 0.f};

    for (int seg = 0; seg < 2; ++seg) {
        const float* Aseg = seg ? A1 : A0;
        const float* Bseg = seg ? B1 : B0;
        for (int kc = 0; kc < K; kc += KC) {
            // ---- stage A: 32 rows x KC floats (scaled by 1/deg for the agg segment) ----
            #pragma unroll
            for (int i = 0; i < 2; ++i) {
                int idx = tid + (i << 8);            // 0..511
                int r   = idx >> 4;                  // 0..31
                int c   = (idx & 15) << 2;           // 0..60
                int row = m0 + r;
                float4 v = make_float4(0.f, 0.f, 0.f, 0.f);
                float  sc = 1.0f;
                if (row < M) {
                    const float* ap = Aseg + (size_t)row * K + kc + c;
                    v = *(const float4*)ap;
                    if (seg == 0) sc = inv[row];
                    if (kc + KC < K) __builtin_prefetch(ap + KC, 0, 0);
                }
                As[r][c + 0] = v.x * sc;
                As[r][c + 1] = v.y * sc;
                As[r][c + 2] = v.z * sc;
                As[r][c + 3] = v.w * sc;
            }
            // ---- stage B: KC k-rows x 64 cols (coalesced from transposed weights) ----
            #pragma unroll
            for (int i = 0; i < 4; ++i) {
                int idx = tid + (i << 8);            // 0..1023
                int r   = idx >> 4;                  // 0..63
                int c   = (idx & 15) << 2;           // 0..60
                const float4 v = *(const float4*)(Bseg + (size_t)(kc + r) * OutC + n0 + c);
                Bs[r][c + 0] = v.x;
                Bs[r][c + 1] = v.y;
                Bs[r][c + 2] = v.z;
                Bs[r][c + 3] = v.w;
            }
            __syncthreads();

            // ---- WMMA over the chunk ----
            #pragma unroll
            for (int kk = 0; kk < KC; kk += 4) {
                const int kb = kk + (hh << 1);
                v2f a, b;
                a.x = As[(wrow << 4) + ln][kb];
                a.y = As[(wrow << 4) + ln][kb + 1];
                b.x = Bs[kb][(wcol << 4) + ln];
                b.y = Bs[kb + 1][(wcol << 4) + ln];
                acc = __builtin_amdgcn_wmma_f32_16x16x4_f32(
                    /*neg_a=*/false, a, /*neg_b=*/false, b,
                    /*c_mod=*/(short)0, acc, /*reuse_a=*/false, /*reuse_b=*/false);
            }
            __syncthreads();
        }
    }

    // ---- epilogue: bias (+relu), store ----
    const int n = n0 + (wcol << 4) + ln;
    const float bv = bias[n];
    #pragma unroll
    for (int r = 0; r < 8; ++r) {
        int row = m0 + (wrow << 4) + r + (hh << 3);
        if (row < M) {
            float v = acc[r] + bv;
            if (do_relu) v = fmaxf(v, 0.0f);
            out[(size_t)row * OutC + n] = v;
        }
    }
}

// ---------------- global mean pool ----------------
__global__ void gcnt_kernel(const int* __restrict__ batch, float* __restrict__ gcnt, int N) {
    int i = blockIdx.x * blockDim.x + threadIdx.x;
    if (i < N) atomicAdd(&gcnt[batch[i]], 1.0f);
}

__global__ void pool_accum_kernel(const float* __restrict__ h, const int* __restrict__ batch,
                                  float* __restrict__ gsum, int N) {
    int t = blockIdx.x * blockDim.x + threadIdx.x;
    if (t < N * OUT_CH) {
        int i = t >> 7;          // / OUT_CH (==128)
        int c = t & (OUT_CH - 1);
        atomicAdd(&gsum[(size_t)batch[i] * OUT_CH + c], h[t]);
    }
}

__global__ void pool_final_kernel(const float* __restrict__ gsum, const float* __restrict__ gcnt,
                                  float* __restrict__ out) {
    int t = blockIdx.x * blockDim.x + threadIdx.x;
    if (t < N_GRAPHS * OUT_CH) {
        int g = t >> 7;
        out[t] = gsum[t] / fmaxf(gcnt[g], 1.0f);
    }
}

extern "C" void kernel_launch(void* const* d_in, const int* in_sizes, int n_in,
                              void* d_out, int out_size, void* d_ws, size_t ws_size,
                              hipStream_t stream) {
    const float* x     = (const float*)d_in[0];
    const int*   ei    = (const int*)d_in[1];
    const int*   batch = (const int*)d_in[2];
    const float* W1l   = (const float*)d_in[3];
    const float* b1l   = (const float*)d_in[4];
    const float* W1r   = (const float*)d_in[5];
    const float* W2l   = (const float*)d_in[6];
    const float* b2l   = (const float*)d_in[7];
    const float* W2r   = (const float*)d_in[8];

    const int N = in_sizes[0] / IN_CH;
    const int E = in_sizes[1] / 2;
    const int* src = ei;
    const int* dst = ei + E;

    float* out = (float*)d_out;
    float* h2  = out + N_GRAPHS * OUT_CH;   // second tuple element lives in d_out

    // ---- workspace carve ----
    float* p    = (float*)d_ws;
    float* cnt  = p;  p += N;
    float* inv  = p;  p += N;
    float* agg1 = p;  p += (size_t)N * IN_CH;
    float* h1   = p;  p += (size_t)N * HID_CH;
    float* agg2 = p;  p += (size_t)N * HID_CH;
    float* Wt1l = p;  p += IN_CH * HID_CH;
    float* Wt1r = p;  p += IN_CH * HID_CH;
    float* Wt2l = p;  p += HID_CH * OUT_CH;
    float* Wt2r = p;  p += HID_CH * OUT_CH;
    float* gsum = p;  p += N_GRAPHS * OUT_CH;
    float* gcnt = p;  p += N_GRAPHS;

    // ---- zero the accumulator buffers (graph-capturable) ----
    hipMemsetAsync(cnt,  0, (size_t)N * sizeof(float), stream);
    hipMemsetAsync(agg1, 0, (size_t)N * IN_CH * sizeof(float), stream);
    hipMemsetAsync(agg2, 0, (size_t)N * HID_CH * sizeof(float), stream);
    hipMemsetAsync(gsum, 0, (size_t)N_GRAPHS * OUT_CH * sizeof(float), stream);
    hipMemsetAsync(gcnt, 0, (size_t)N_GRAPHS * sizeof(float), stream);

    const int B = 256;

    // degrees (shared by both layers) and reciprocals
    deg_kernel<<<(E + B - 1) / B, B, 0, stream>>>(dst, cnt, E);
    inv_kernel<<<(N + B - 1) / B, B, 0, stream>>>(cnt, inv, N);

    // pre-transpose weights to [K][OutC] for coalesced B staging
    transpose_kernel<<<(HID_CH * IN_CH + B - 1) / B, B, 0, stream>>>(W1l, Wt1l, HID_CH, IN_CH);
    transpose_kernel<<<(HID_CH * IN_CH + B - 1) / B, B, 0, stream>>>(W1r, Wt1r, HID_CH, IN_CH);
    transpose_kernel<<<(OUT_CH * HID_CH + B - 1) / B, B, 0, stream>>>(W2l, Wt2l, OUT_CH, HID_CH);
    transpose_kernel<<<(OUT_CH * HID_CH + B - 1) / B, B, 0, stream>>>(W2r, Wt2r, OUT_CH, HID_CH);

    // ---- layer 1 ----
    {
        long long tot = (long long)E * (IN_CH / 4);
        scatter_kernel<<<(int)((tot + B - 1) / B), B, 0, stream>>>(x, src, dst, agg1, E, IN_CH);
        dim3 grid((N + 31) / 32, HID_CH / 64);
        sage_gemm_kernel<<<grid, B, 0, stream>>>(agg1, x, Wt1l, Wt1r, b1l, inv,
                                                 h1, N, IN_CH, HID_CH, /*relu=*/1);
    }

    // ---- layer 2 (writes h directly into d_out tail) ----
    {
        long long tot = (long long)E * (HID_CH / 4);
        scatter_kernel<<<(int)((tot + B - 1) / B), B, 0, stream>>>(h1, src, dst, agg2, E, HID_CH);
        dim3 grid((N + 31) / 32, OUT_CH / 64);
        sage_gemm_kernel<<<grid, B, 0, stream>>>(agg2, h1, Wt2l, Wt2r, b2l, inv,
                                                 h2, N, HID_CH, OUT_CH, /*relu=*/0);
    }

    // ---- global mean pool into d_out head ----
    gcnt_kernel<<<(N + B - 1) / B, B, 0, stream>>>(batch, gcnt, N);
    pool_accum_kernel<<<((long long)N * OUT_CH + B - 1) / B, B, 0, stream>>>(h2, batch, gsum, N);
    pool_final_kernel<<<(N_GRAPHS * OUT_CH + B - 1) / B, B, 0, stream>>>(gsum, gcnt, out);
}